// Net_9105330668226
// MI455X (gfx1250) — compile-verified
//
#include <hip/hip_runtime.h>
#include <hip/hip_bf16.h>
#include <math.h>

// ---------------------------------------------------------------------------
// Shapes from the reference
// ---------------------------------------------------------------------------
enum { B_ = 256, N_ = 128, L_ = 2048, NF_ = 75, AC_ = 3, HID_ = 256,
       DIM_ = 128, NHEAD_ = 8, FPD_ = 679, NWORD_ = 8600 };

typedef __attribute__((ext_vector_type(2))) float v2f;
typedef __attribute__((ext_vector_type(8))) float v8f;

#define GFLAG_ACC  1
#define GFLAG_RELU 2

// ---------------------------------------------------------------------------
// FAST fp32 WMMA GEMM (no edge guards):
//   requires M%128==0, Nc%32==0, K%4==0, lda even, 8B-aligned base pointers.
// One wave -> 32x32 C tile (2x2 subtiles of V_WMMA_F32_16X16X4_F32).
// Per k-step: 2x global_load_b64 (A) + 4x b32 (W) + 4 WMMA = 8192 FLOP.
// ---------------------------------------------------------------------------
__global__ __launch_bounds__(128)
void wmma_gemm_f32_fast(const float* __restrict__ A, long long strideA, int lda,
                        const float* __restrict__ W, long long strideW,
                        const float* __restrict__ bias,
                        float* __restrict__ C, long long strideC, int ldc,
                        int K, int Nc, int flags)
{
    const int wave = threadIdx.x >> 5;
    const int lane = threadIdx.x & 31;
    const int half = lane >> 4;
    const int l15  = lane & 15;
    const int tileM = (blockIdx.y * 4 + wave) * 32;
    const int tileN = blockIdx.x * 32;

    A += (long long)blockIdx.z * strideA;
    W += (long long)blockIdx.z * strideW;
    C += (long long)blockIdx.z * strideC;

    // A: lane reads row (tileM + mi*16 + l15), K-pair (k+2*half, k+2*half+1)
    const float* pA0 = A + (long long)(tileM + l15) * lda + 2 * half;
    const float* pA1 = pA0 + (long long)16 * lda;
    // W: lane reads rows (k+2*half, k+2*half+1), col (tileN + ni*16 + l15)
    const float* pW0 = W + (long long)(2 * half) * Nc + tileN + l15;
    const float* pW1 = pW0 + 16;

    v8f acc00 = {}, acc01 = {}, acc10 = {}, acc11 = {};
    if (flags & GFLAG_ACC) {
#pragma unroll
        for (int r = 0; r < 8; ++r) {
            const long long r0 = (long long)(tileM + r + 8 * half) * ldc;
            const long long r1 = r0 + (long long)16 * ldc;
            acc00[r] = C[r0 + tileN + l15];
            acc01[r] = C[r0 + tileN + 16 + l15];
            acc10[r] = C[r1 + tileN + l15];
            acc11[r] = C[r1 + tileN + 16 + l15];
        }
    }

#pragma unroll 4
    for (int k = 0; k < K; k += 4) {
        v2f a0 = *(const v2f*)(pA0 + k);
        v2f a1 = *(const v2f*)(pA1 + k);
        const float* w0 = pW0 + (long long)k * Nc;
        const float* w1 = pW1 + (long long)k * Nc;
        v2f b0, b1;
        b0.x = w0[0];  b0.y = w0[Nc];
        b1.x = w1[0];  b1.y = w1[Nc];
        acc00 = __builtin_amdgcn_wmma_f32_16x16x4_f32(false, a0, false, b0, (short)0, acc00, false, false);
        acc01 = __builtin_amdgcn_wmma_f32_16x16x4_f32(false, a0, false, b1, (short)0, acc01, false, false);
        acc10 = __builtin_amdgcn_wmma_f32_16x16x4_f32(false, a1, false, b0, (short)0, acc10, false, false);
        acc11 = __builtin_amdgcn_wmma_f32_16x16x4_f32(false, a1, false, b1, (short)0, acc11, false, false);
    }

    float bb0 = 0.f, bb1 = 0.f;
    if (bias) { bb0 = bias[tileN + l15]; bb1 = bias[tileN + 16 + l15]; }
    const bool relu = (flags & GFLAG_RELU) != 0;
#pragma unroll
    for (int r = 0; r < 8; ++r) {
        const long long r0 = (long long)(tileM + r + 8 * half) * ldc;
        const long long r1 = r0 + (long long)16 * ldc;
        float v00 = acc00[r] + bb0, v01 = acc01[r] + bb1;
        float v10 = acc10[r] + bb0, v11 = acc11[r] + bb1;
        if (relu) {
            v00 = fmaxf(v00, 0.f); v01 = fmaxf(v01, 0.f);
            v10 = fmaxf(v10, 0.f); v11 = fmaxf(v11, 0.f);
        }
        C[r0 + tileN + l15]      = v00;
        C[r0 + tileN + 16 + l15] = v01;
        C[r1 + tileN + l15]      = v10;
        C[r1 + tileN + 16 + l15] = v11;
    }
}

// ---------------------------------------------------------------------------
// EDGE fp32 WMMA GEMM (clamped addresses + cndmask selects; any M/K/Nc).
// One wave -> one 16x16 tile. Bitwise '&' on guards avoids saveexec branches
// so EXEC stays all-ones at the WMMA.
// ---------------------------------------------------------------------------
__global__ __launch_bounds__(128)
void wmma_gemm_f32_edge(const float* __restrict__ A, long long strideA, int lda,
                        const float* __restrict__ W, long long strideW,
                        const float* __restrict__ bias,
                        float* __restrict__ C, long long strideC, int ldc,
                        int M, int K, int Nc, int flags)
{
    const int wave = threadIdx.x >> 5;
    const int lane = threadIdx.x & 31;
    const int half = lane >> 4;
    const int l15  = lane & 15;
    const int tileM = (blockIdx.y * 4 + wave) * 16;
    const int tileN = blockIdx.x * 16;

    A += (long long)blockIdx.z * strideA;
    W += (long long)blockIdx.z * strideW;
    C += (long long)blockIdx.z * strideC;

    const int  arow  = tileM + l15;
    const int  arowc = arow < M ? arow : (M - 1);
    const bool arowv = arow < M;
    const int  col   = tileN + l15;
    const int  colc  = col < Nc ? col : (Nc - 1);
    const bool colv  = col < Nc;

    v8f acc = {};
    if (flags & GFLAG_ACC) {
#pragma unroll
        for (int r = 0; r < 8; ++r) {
            int row  = tileM + r + 8 * half;
            int rowc = row < M ? row : (M - 1);
            float v = C[(long long)rowc * ldc + colc];
            acc[r] = (bool)((row < M) & colv) ? v : 0.0f;
        }
    }

    const float* pA = A + (long long)arowc * lda;
    for (int k = 0; k < K; k += 4) {
        const int  k0  = k + 2 * half;
        const int  k1  = k0 + 1;
        const int  k0c = k0 < K ? k0 : (K - 1);
        const int  k1c = k1 < K ? k1 : (K - 1);
        const bool v0  = arowv & (k0 < K);
        const bool v1  = arowv & (k1 < K);
        const bool w0v = colv & (k0 < K);
        const bool w1v = colv & (k1 < K);

        float a0 = pA[k0c];
        float a1 = pA[k1c];
        float b0 = W[(long long)k0c * Nc + colc];
        float b1 = W[(long long)k1c * Nc + colc];

        v2f av, bv;
        av.x = v0  ? a0 : 0.0f;
        av.y = v1  ? a1 : 0.0f;
        bv.x = w0v ? b0 : 0.0f;
        bv.y = w1v ? b1 : 0.0f;

        acc = __builtin_amdgcn_wmma_f32_16x16x4_f32(
            false, av, false, bv, (short)0, acc, false, false);
    }

    const float bb = (bias != nullptr && colv) ? bias[col] : 0.0f;
#pragma unroll
    for (int r = 0; r < 8; ++r) {
        int row = tileM + r + 8 * half;
        if ((bool)((row < M) & colv)) {
            float v = acc[r] + bb;
            if (flags & GFLAG_RELU) v = fmaxf(v, 0.0f);
            C[(long long)row * ldc + col] = v;
        }
    }
}

// ---------------------------------------------------------------------------
// Masked multi-head global attention pooling. One block per molecule.
// pooled[b, h*HID + f] = sum_n softmax_n(scores)[n,h] * Vh[b,n,f]
// ---------------------------------------------------------------------------
__global__ __launch_bounds__(256)
void attn_pool(const float* __restrict__ Vh, const float* __restrict__ attW,
               const int* __restrict__ mol_size, float* __restrict__ pooled)
{
    __shared__ float sc[N_ * NHEAD_];
    __shared__ float cmax[NHEAD_], csum[NHEAD_];
    const int b = blockIdx.x;
    const int tid = threadIdx.x;
    const float* V = Vh + (long long)b * N_ * HID_;
    const int ms = mol_size[b];

    for (int idx = tid; idx < N_ * NHEAD_; idx += 256) {
        int n = idx >> 3, h = idx & 7;
        float s = 0.f;
        const float* vr = V + n * HID_;
        for (int f = 0; f < HID_; ++f) s += vr[f] * attW[f * NHEAD_ + h];
        sc[idx] = (n < ms) ? s : -1e9f;
    }
    __syncthreads();
    if (tid < NHEAD_) {
        float m = -1e30f;
        for (int n = 0; n < N_; ++n) m = fmaxf(m, sc[n * NHEAD_ + tid]);
        float s = 0.f;
        for (int n = 0; n < N_; ++n) s += expf(sc[n * NHEAD_ + tid] - m);
        cmax[tid] = m; csum[tid] = s;
    }
    __syncthreads();
    for (int idx = tid; idx < N_ * NHEAD_; idx += 256) {
        int h = idx & 7;
        sc[idx] = expf(sc[idx] - cmax[h]) / csum[h];
    }
    __syncthreads();
    for (int idx = tid; idx < NHEAD_ * HID_; idx += 256) {
        int h = idx >> 8, f = idx & 255;
        float s = 0.f;
        for (int n = 0; n < N_; ++n) s += sc[n * NHEAD_ + h] * V[n * HID_ + f];
        pooled[(long long)b * (NHEAD_ * HID_) + idx] = s;
    }
}

// ---------------------------------------------------------------------------
// Cross-attention max:  out[b,:] = max_l tanh(h . T[seq[b,l]]) * T[seq[b,l]]
// One block per b; each wave owns a strided slice of l; lanes split DIM as
// float4; dot reduced with 5x shfl_xor (wave32).
// ---------------------------------------------------------------------------
__global__ __launch_bounds__(256)
void cross_attn_max(const float* __restrict__ H, const float* __restrict__ T,
                    const int* __restrict__ seq, float* __restrict__ out)
{
    __shared__ __align__(16) float hsh[DIM_];
    __shared__ float wmax[8][DIM_];
    const int b = blockIdx.x;
    const int tid = threadIdx.x;
    const int wave = tid >> 5, lane = tid & 31;

    if (tid < DIM_) hsh[tid] = H[b * DIM_ + tid];
    __syncthreads();

    const float4 hv = reinterpret_cast<const float4*>(hsh)[lane];
    const float4* T4 = reinterpret_cast<const float4*>(T);
    float m0 = -1e30f, m1 = -1e30f, m2 = -1e30f, m3 = -1e30f;

    for (int l = wave; l < L_; l += 8) {
        int idx = seq[(long long)b * L_ + l];
        float4 t = T4[(long long)idx * (DIM_ / 4) + lane];
        float p = t.x * hv.x + t.y * hv.y + t.z * hv.z + t.w * hv.w;
#pragma unroll
        for (int off = 16; off > 0; off >>= 1) p += __shfl_xor(p, off);
        float w = tanhf(p);
        m0 = fmaxf(m0, w * t.x);
        m1 = fmaxf(m1, w * t.y);
        m2 = fmaxf(m2, w * t.z);
        m3 = fmaxf(m3, w * t.w);
    }
    wmax[wave][lane * 4 + 0] = m0;
    wmax[wave][lane * 4 + 1] = m1;
    wmax[wave][lane * 4 + 2] = m2;
    wmax[wave][lane * 4 + 3] = m3;
    __syncthreads();
    if (tid < DIM_) {
        float m = wmax[0][tid];
        for (int w2 = 1; w2 < 8; ++w2) m = fmaxf(m, wmax[w2][tid]);
        out[b * DIM_ + tid] = m;
    }
}

// ---------------------------------------------------------------------------
// cat = [X | Y] per row
// ---------------------------------------------------------------------------
__global__ void concat2(const float* __restrict__ X, const float* __restrict__ Y,
                        float* __restrict__ C, int rows, int D)
{
    int i = blockIdx.x * blockDim.x + threadIdx.x;
    int total = rows * 2 * D;
    if (i < total) {
        int b = i / (2 * D), j = i % (2 * D);
        C[i] = (j < D) ? X[b * D + j] : Y[b * D + (j - D)];
    }
}

// ---------------------------------------------------------------------------
// Final heads + argmax + softmax score; writes all 4 tuple outputs.
// ---------------------------------------------------------------------------
__global__ void head_kernel(const float* __restrict__ x1, const float* __restrict__ x2,
                            const float* __restrict__ W1, const float* __restrict__ b1,
                            const float* __restrict__ W2, const float* __restrict__ b2,
                            const int* __restrict__ label, float* __restrict__ out)
{
    int b = blockIdx.x * blockDim.x + threadIdx.x;
    if (b >= B_) return;
    float p0 = b1[0] + b2[0], p1 = b1[1] + b2[1];
    const float* c1 = x1 + b * 2 * DIM_;
    const float* c2 = x2 + b * 2 * DIM_;
    for (int k = 0; k < 2 * DIM_; ++k) {
        p0 += c1[k] * W1[k * 2 + 0] + c2[k] * W2[k * 2 + 0];
        p1 += c1[k] * W1[k * 2 + 1] + c2[k] * W2[k * 2 + 1];
    }
    out[b * 2 + 0] = p0;
    out[b * 2 + 1] = p1;
    out[2 * B_ + b] = (float)label[b];          // label passthrough
    out[3 * B_ + b] = (p1 > p0) ? 1.0f : 0.0f;  // argmax (tie -> 0)
    float mm = fmaxf(p0, p1);
    float e0 = expf(p0 - mm), e1 = expf(p1 - mm);
    out[4 * B_ + b] = e1 / (e0 + e1);           // softmax[:,1]
}

// ---------------------------------------------------------------------------
// Host-side orchestration
// ---------------------------------------------------------------------------
extern "C" void kernel_launch(void* const* d_in, const int* in_sizes, int n_in,
                              void* d_out, int out_size, void* d_ws, size_t ws_size,
                              hipStream_t stream)
{
    const float* V        = (const float*)d_in[0];
    const float* Aadj     = (const float*)d_in[1];
    const float* fp       = (const float*)d_in[2];
    const int*   mol_size = (const int*)  d_in[3];
    const int*   seq      = (const int*)  d_in[4];
    const int*   label    = (const int*)  d_in[5];
    const float* gc_W1 = (const float*)d_in[6];
    const float* gc_b1 = (const float*)d_in[7];
    const float* gc_W2 = (const float*)d_in[8];
    const float* gc_b2 = (const float*)d_in[9];
    const float* gc_W3 = (const float*)d_in[10];
    const float* gc_b3 = (const float*)d_in[11];
    const float* fp_W1 = (const float*)d_in[12];
    const float* fp_b1 = (const float*)d_in[13];
    const float* fp_W2 = (const float*)d_in[14];
    const float* fp_b2 = (const float*)d_in[15];
    const float* att_W = (const float*)d_in[16];
    const float* ro_W  = (const float*)d_in[17];
    const float* ro_b  = (const float*)d_in[18];
    const float* emb   = (const float*)d_in[19];
    const float* mfp_W = (const float*)d_in[20];
    const float* mfp_b = (const float*)d_in[21];
    const float* pfp_W = (const float*)d_in[22];
    const float* pfp_b = (const float*)d_in[23];
    const float* mv_W  = (const float*)d_in[24];
    const float* mv_b  = (const float*)d_in[25];
    const float* pv_W  = (const float*)d_in[26];
    const float* pv_b  = (const float*)d_in[27];
    const float* mlp1_W = (const float*)d_in[28];
    const float* mlp1_b = (const float*)d_in[29];
    const float* mlp2_W = (const float*)d_in[30];
    const float* mlp2_b = (const float*)d_in[31];
    const float* out1_W = (const float*)d_in[32];
    const float* out1_b = (const float*)d_in[33];
    const float* out2_W = (const float*)d_in[34];
    const float* out2_b = (const float*)d_in[35];
    float* out = (float*)d_out;
    (void)in_sizes; (void)n_in; (void)out_size; (void)ws_size;

    // Workspace carve-out (floats)
    float* ws = (float*)d_ws;
    size_t off = 0;
    auto alloc = [&](size_t n) { float* p = ws + off; off += n; return p; };
    float* FPa    = alloc((size_t)B_ * DIM_);
    float* FP     = alloc((size_t)B_ * DIM_);
    float* Vh_a   = alloc((size_t)B_ * N_ * HID_);
    float* Vh_b   = alloc((size_t)B_ * N_ * HID_);
    float* AVbuf  = alloc((size_t)B_ * N_ * HID_);   // reused per channel
    float* pooled = alloc((size_t)B_ * NHEAD_ * HID_);
    float* Vr     = alloc((size_t)B_ * DIM_);
    float* T1     = alloc((size_t)NWORD_ * DIM_);
    float* T2     = alloc((size_t)NWORD_ * DIM_);
    float* h1     = alloc((size_t)B_ * DIM_);
    float* h2     = alloc((size_t)B_ * DIM_);
    float* it1    = alloc((size_t)B_ * DIM_);
    float* it2    = alloc((size_t)B_ * DIM_);
    float* x1a    = alloc((size_t)B_ * 2 * DIM_);
    float* x1b    = alloc((size_t)B_ * 2 * DIM_);
    float* x2a    = alloc((size_t)B_ * 2 * DIM_);
    float* x2b    = alloc((size_t)B_ * 2 * DIM_);

    const dim3 blk(128);
    auto gemm = [&](const float* Am, long long sA, int lda,
                    const float* Wm, long long sW, const float* bias,
                    float* Cm, long long sC, int ldc,
                    int M, int K, int Nc, int flags, int batch) {
        const bool fastok = (M % 128 == 0) && (Nc % 32 == 0) &&
                            (K % 4 == 0) && (lda % 2 == 0);
        if (fastok) {
            dim3 grid(Nc / 32, M / 128, batch);
            wmma_gemm_f32_fast<<<grid, blk, 0, stream>>>(Am, sA, lda, Wm, sW, bias,
                                                         Cm, sC, ldc, K, Nc, flags);
        } else {
            dim3 grid((Nc + 15) / 16, (M + 63) / 64, batch);
            wmma_gemm_f32_edge<<<grid, blk, 0, stream>>>(Am, sA, lda, Wm, sW, bias,
                                                         Cm, sC, ldc, M, K, Nc, flags);
        }
    };

    // --- fingerprint chain ---------------------------------------------------
    gemm(fp, 0, FPD_, fp_W1, 0, fp_b1, FPa, 0, DIM_, B_, FPD_, DIM_, GFLAG_RELU, 1);
    gemm(FPa, 0, DIM_, fp_W2, 0, fp_b2, FP, 0, DIM_, B_, DIM_, DIM_, GFLAG_RELU, 1);

    // --- graph conv blocks ---------------------------------------------------
    // Vout = relu( sum_c (A[:,c] @ Vin) @ W[c] + b ), accumulated over c in C.
    auto gconv = [&](const float* Vin, int Fin, const float* Wc, const float* bc,
                     float* Vout) {
        for (int c = 0; c < AC_; ++c) {
            // AV[b] = A[b,c] (128x128) @ Vin[b] (128xFin)
            gemm(Aadj + (size_t)c * N_ * N_, (long long)AC_ * N_ * N_, N_,
                 Vin, (long long)N_ * Fin, nullptr,
                 AVbuf, (long long)N_ * Fin, Fin,
                 N_, N_, Fin, 0, B_);
            // Vout (+)= AV @ W[c]  (relu+bias on last channel)
            int fl = (c == 0 ? 0 : GFLAG_ACC) | (c == AC_ - 1 ? GFLAG_RELU : 0);
            gemm(AVbuf, 0, Fin, Wc + (size_t)c * Fin * HID_, 0,
                 (c == AC_ - 1 ? bc : nullptr),
                 Vout, 0, HID_, B_ * N_, Fin, HID_, fl, 1);
        }
    };
    gconv(V,    NF_,  gc_W1, gc_b1, Vh_a);
    gconv(Vh_a, HID_, gc_W2, gc_b2, Vh_b);
    gconv(Vh_b, HID_, gc_W3, gc_b3, Vh_a);

    // --- masked attention pooling + readout ----------------------------------
    attn_pool<<<B_, 256, 0, stream>>>(Vh_a, att_W, mol_size, pooled);
    gemm(pooled, 0, NHEAD_ * HID_, ro_W, 0, ro_b, Vr, 0, DIM_,
         B_, NHEAD_ * HID_, DIM_, GFLAG_RELU, 1);

    // --- protein tables: hs = relu(emb @ pW + pb) gathered later -------------
    gemm(emb, 0, DIM_, pfp_W, 0, pfp_b, T1, 0, DIM_, NWORD_, DIM_, DIM_, GFLAG_RELU, 1);
    gemm(emb, 0, DIM_, pv_W,  0, pv_b,  T2, 0, DIM_, NWORD_, DIM_, DIM_, GFLAG_RELU, 1);

    // --- query projections ---------------------------------------------------
    gemm(FP, 0, DIM_, mfp_W, 0, mfp_b, h1, 0, DIM_, B_, DIM_, DIM_, GFLAG_RELU, 1);
    gemm(Vr, 0, DIM_, mv_W,  0, mv_b,  h2, 0, DIM_, B_, DIM_, DIM_, GFLAG_RELU, 1);

    // --- ragged cross-attention max ------------------------------------------
    cross_attn_max<<<B_, 256, 0, stream>>>(h1, T1, seq, it1);
    cross_attn_max<<<B_, 256, 0, stream>>>(h2, T2, seq, it2);

    // --- concat + MLP stacks -------------------------------------------------
    concat2<<<(B_ * 2 * DIM_ + 255) / 256, 256, 0, stream>>>(FP, it1, x1a, B_, DIM_);
    concat2<<<(B_ * 2 * DIM_ + 255) / 256, 256, 0, stream>>>(Vr, it2, x2a, B_, DIM_);

    float* a1 = x1a; float* b1p = x1b;
    float* a2 = x2a; float* b2p = x2b;
    for (int i = 0; i < 3; ++i) {
        gemm(a1, 0, 2 * DIM_, mlp1_W + (size_t)i * 2 * DIM_ * 2 * DIM_, 0,
             mlp1_b + (size_t)i * 2 * DIM_, b1p, 0, 2 * DIM_,
             B_, 2 * DIM_, 2 * DIM_, GFLAG_RELU, 1);
        gemm(a2, 0, 2 * DIM_, mlp2_W + (size_t)i * 2 * DIM_ * 2 * DIM_, 0,
             mlp2_b + (size_t)i * 2 * DIM_, b2p, 0, 2 * DIM_,
             B_, 2 * DIM_, 2 * DIM_, GFLAG_RELU, 1);
        float* t;
        t = a1; a1 = b1p; b1p = t;
        t = a2; a2 = b2p; b2p = t;
    }

    // --- heads + outputs -----------------------------------------------------
    head_kernel<<<(B_ + 63) / 64, 64, 0, stream>>>(a1, a2, out1_W, out1_b,
                                                   out2_W, out2_b, label, out);
}